// Softsplat_Count_33887291966095
// MI455X (gfx1250) — compile-verified
//
#include <hip/hip_runtime.h>

// ---------------- Problem constants (from reference) ----------------
constexpr int BATCH = 16;
constexpr int IMG_H = 720;
constexpr int IMG_W = 1280;

// ---------------- Tiling configuration ----------------
// One block = one 160x144 source tile of one batch image (exact tiling:
// 1280 = 8*160, 720 = 5*144). LDS accumulator covers tile + 32px halo:
// flow ~ N(0,10) so |disp|<=32 covers ~3.2 sigma per axis (~99.7% of splats);
// the rest fall back to direct global atomics.
constexpr int TW   = 160;
constexpr int TH   = 144;
constexpr int RAD  = 32;
constexpr int EXPW = TW + 2 * RAD;   // 224
constexpr int EXPH = TH + 2 * RAD;   // 208  -> acc = 224*208*4B = 186,368 B LDS
constexpr int NTHR = 1024;           // 32 wave32 waves -> fills one WGP

// Flow staging: double-buffered async global->LDS copy, CH rows per chunk,
// both flow planes. 2*16*160 = 5120 dwords per chunk -> exactly 5 async b32
// loads per thread per chunk (uniform => s_wait_asynccnt immediates are exact).
constexpr int CH          = 16;
constexpr int NCH         = TH / CH;            // 9 chunks
constexpr int CHUNK_WORDS = 2 * CH * TW;        // 5120
constexpr int LPT         = CHUNK_WORDS / NTHR; // 5
static_assert(CHUNK_WORDS % NTHR == 0, "chunk must divide evenly");
static_assert(LPT == 5, "s_wait_asynccnt immediate below assumes 5 loads/chunk");

// ---- Pinned CDNA5 atomics (guarantee single-instruction hardware forms) ----

// LDS float atomic add, no return: DS_ADD_F32 (ISA 15.15 op 21).
// Takes the 32-bit LDS byte address (low 32 bits of the generic pointer).
__device__ __forceinline__ void lds_atomic_add(const float* p, float w) {
  const unsigned a = (unsigned)(uintptr_t)p;
  asm volatile("ds_add_f32 %0, %1" :: "v"(a), "v"(w) : "memory");
}

// Global float atomic add, no return, device scope: GLOBAL_ATOMIC_ADD_F32
// (ISA 15.18.3 op 86). Tracked by STOREcnt; S_ENDPGM's implicit wait-idle
// covers completion at kernel end.
__device__ __forceinline__ void global_atomic_add(float* p, float w) {
  asm volatile("global_atomic_add_f32 %0, %1, off scope:SCOPE_DEV"
               :: "v"((unsigned long long)(uintptr_t)p), "v"(w)
               : "memory");
}

// Issue one chunk of flow (both planes) as async global->LDS b32 copies.
// sbase = &flow[b, 0, rowBase, tx0] (block-uniform -> SGPR pair).
// Per-lane dword offset selects plane/row/col. Tracked by ASYNCcnt.
__device__ __forceinline__ void issue_chunk(const float* __restrict__ sbase,
                                            float* fbuf, int bufsel, int tid) {
#pragma unroll
  for (int k = 0; k < LPT; ++k) {
    const int i   = tid + k * NTHR;            // 0..5119
    const int p   = (i >= CH * TW) ? 1 : 0;    // flow plane (x-disp / y-disp)
    const int rem = i - p * (CH * TW);
    const int r   = rem / TW;
    const int x   = rem - r * TW;
    const unsigned voff =
        (unsigned)((p * IMG_H * IMG_W + r * IMG_W + x) * 4);
    const unsigned lds =
        (unsigned)(uintptr_t)(&fbuf[bufsel * CHUNK_WORDS + i]);
    asm volatile("global_load_async_to_lds_b32 %0, %1, %2"
                 :
                 : "v"(lds), "v"(voff), "s"(sbase)
                 : "memory");
  }
}

// Add weight w at output pixel (cx,cy): LDS tile atomic if inside the
// expanded tile, else direct global atomic fallback (rare tail).
__device__ __forceinline__ void splat_one(int cx, int cy, float w,
                                          int ox, int oy,
                                          float* acc, float* outb) {
  if ((unsigned)cx < (unsigned)IMG_W && (unsigned)cy < (unsigned)IMG_H) {
    const int lx = cx - ox;
    const int ly = cy - oy;
    if ((unsigned)lx < (unsigned)EXPW && (unsigned)ly < (unsigned)EXPH) {
      lds_atomic_add(&acc[ly * EXPW + lx], w);       // ds_add_f32
    } else {
      global_atomic_add(&outb[cy * IMG_W + cx], w);  // global_atomic_add_f32
    }
  }
}

__global__ __launch_bounds__(NTHR, 1)
void softsplat_count_kernel(const float* __restrict__ flow,
                            float* __restrict__ out) {
  __shared__ float acc[EXPW * EXPH];        // ~186 KB tile accumulator
  __shared__ float fbuf[2 * CHUNK_WORDS];   // ~40 KB double-buffered flow stage

  const int tid = threadIdx.x;
  const int b   = blockIdx.z;
  const int tx0 = blockIdx.x * TW;
  const int ty0 = blockIdx.y * TH;
  const int ox  = tx0 - RAD;
  const int oy  = ty0 - RAD;

  const float* flowb = flow + (size_t)b * 2 * IMG_H * IMG_W;
  float*       outb  = out  + (size_t)b * IMG_H * IMG_W;

  // Kick off chunk 0 DMA; zero the accumulator while it is in flight.
  issue_chunk(flowb + (size_t)ty0 * IMG_W + tx0, fbuf, 0, tid);
  for (int i = tid; i < EXPW * EXPH; i += NTHR) acc[i] = 0.0f;
  __syncthreads();

  for (int c = 0; c < NCH; ++c) {
    if (c + 1 < NCH) {
      // Prefetch next chunk, then retire the current one:
      // outstanding = 5 (cur) + 5 (next); wait <=5 => current chunk landed.
      issue_chunk(flowb + (size_t)(ty0 + (c + 1) * CH) * IMG_W + tx0,
                  fbuf, (c + 1) & 1, tid);
      asm volatile("s_wait_asynccnt 5" ::: "memory");
    } else {
      asm volatile("s_wait_asynccnt 0" ::: "memory");
    }
    __syncthreads();  // all waves' chunk-c data visible in LDS

    const float* fb      = &fbuf[(c & 1) * CHUNK_WORDS];
    const int    rowBase = ty0 + c * CH;

    for (int i = tid; i < CH * TW; i += NTHR) {
      const int   r  = i / TW;
      const int   x  = i - r * TW;
      const float fx = fb[i];             // plane 0: x displacement
      const float fy = fb[CH * TW + i];   // plane 1: y displacement

      const float txf = (float)(tx0 + x)     + fx;
      const float tyf = (float)(rowBase + r) + fy;
      const float fx0 = floorf(txf);
      const float fy0 = floorf(tyf);
      const int   x0  = (int)fx0;
      const int   y0  = (int)fy0;
      const float ax  = txf - fx0;
      const float ay  = tyf - fy0;

      splat_one(x0,     y0,     (1.0f - ax) * (1.0f - ay), ox, oy, acc, outb);
      splat_one(x0 + 1, y0,     ax          * (1.0f - ay), ox, oy, acc, outb);
      splat_one(x0,     y0 + 1, (1.0f - ax) * ay,          ox, oy, acc, outb);
      splat_one(x0 + 1, y0 + 1, ax          * ay,          ox, oy, acc, outb);
    }
    __syncthreads();  // reads done before this buffer is overwritten again
  }

  // The compiler cannot track the inline-asm ds_add_f32 in its DScnt
  // bookkeeping: each wave retires its own LDS atomics, then the barrier
  // makes every wave's atomics complete before anyone reads acc.
  asm volatile("s_wait_dscnt 0" ::: "memory");
  __syncthreads();

  // Flush expanded tile to global with hardware float atomics; skip exact
  // zeros (most of the 32px halo tail) to cut L2 atomic traffic further.
  for (int i = tid; i < EXPW * EXPH; i += NTHR) {
    const float v = acc[i];
    if (v != 0.0f) {
      const int ly = i / EXPW;
      const int lx = i - ly * EXPW;
      const int gx = ox + lx;
      const int gy = oy + ly;
      if ((unsigned)gx < (unsigned)IMG_W && (unsigned)gy < (unsigned)IMG_H)
        global_atomic_add(&outb[gy * IMG_W + gx], v);
    }
  }
}

// Output accumulates via atomics, and the harness poisons d_out -> zero it.
__global__ void zero_out_kernel(float4* __restrict__ out, int n4) {
  const int i = blockIdx.x * blockDim.x + threadIdx.x;
  if (i < n4) out[i] = make_float4(0.0f, 0.0f, 0.0f, 0.0f);
}

extern "C" void kernel_launch(void* const* d_in, const int* in_sizes, int n_in,
                              void* d_out, int out_size, void* d_ws, size_t ws_size,
                              hipStream_t stream) {
  // setup_inputs order: d_in[0] = img (unused by the reference output),
  //                     d_in[1] = flow [B,2,H,W] float32.
  const float* flow = (const float*)d_in[1];
  float*       out  = (float*)d_out;

  const int n  = BATCH * IMG_H * IMG_W;   // 14,745,600 (16B-divisible)
  const int n4 = n / 4;
  zero_out_kernel<<<(n4 + 255) / 256, 256, 0, stream>>>((float4*)out, n4);

  dim3 grid(IMG_W / TW, IMG_H / TH, BATCH);  // 8 x 5 x 16 tiles
  softsplat_count_kernel<<<grid, NTHR, 0, stream>>>(flow, out);
}